// Orthogonalize_11201274708388
// MI455X (gfx1250) — compile-verified
//
#include <hip/hip_runtime.h>

#define DD 2048
#define MM 512
#define NSTEPS 20

typedef __attribute__((ext_vector_type(2))) float v2f;
typedef __attribute__((ext_vector_type(8))) float v8f;

static __device__ __forceinline__ v8f wmma_f32(v2f a, v2f b, v8f c) {
  // D(16x16) = A(16x4) * B(4x16) + C, full fp32 — V_WMMA_F32_16X16X4_F32
  return __builtin_amdgcn_wmma_f32_16x16x4_f32(
      /*neg_a=*/false, a, /*neg_b=*/false, b,
      /*c_mod=*/(short)0, c, /*reuse_a=*/false, /*reuse_b=*/false);
}

// ---------------------------------------------------------------------------
// G-partial: each wave computes a 32x32 block of (X^T X) over one K-chunk.
// X is row-major [DD, MM].  A = X^T (MxK view), B = X (KxM view).
// Per-lane operand layout (f32 16x16x4 WMMA):
//   lane<16 : row/col = lane,     v0 -> K=0, v1 -> K=1
//   lane>=16: row/col = lane-16,  v0 -> K=2, v1 -> K=3
// ---------------------------------------------------------------------------
__global__ __launch_bounds__(128) void gram_partial_kernel(
    const float* __restrict__ X, float* __restrict__ P, int chunk) {
  const int wave = threadIdx.x >> 5;
  const int lane = threadIdx.x & 31;
  const int tile = blockIdx.x * 4 + wave;       // 0..255  (16x16 grid of 32x32 blocks)
  const int i0 = (tile >> 4) * 32;
  const int j0 = (tile & 15) * 32;
  const int idx = lane & 15;
  const int kb  = (lane >> 4) << 1;             // 0 or 2
  const int k0  = blockIdx.y * chunk;
  float* __restrict__ Pp = P + (size_t)blockIdx.y * MM * MM;

  v8f c00 = {}, c01 = {}, c10 = {}, c11 = {};
  #pragma unroll 4
  for (int k = 0; k < chunk; k += 4) {
    const float* Xk0 = X + (size_t)(k0 + k + kb) * MM;  // row k+kb   of X
    const float* Xk1 = Xk0 + MM;                        // row k+kb+1 of X
    v2f a0, a1, b0, b1;
    a0.x = Xk0[i0 + idx];       a0.y = Xk1[i0 + idx];        // A rows i0..i0+15
    a1.x = Xk0[i0 + 16 + idx];  a1.y = Xk1[i0 + 16 + idx];   // A rows i0+16..i0+31
    b0.x = Xk0[j0 + idx];       b0.y = Xk1[j0 + idx];        // B cols j0..j0+15
    b1.x = Xk0[j0 + 16 + idx];  b1.y = Xk1[j0 + 16 + idx];   // B cols j0+16..j0+31
    c00 = wmma_f32(a0, b0, c00);
    c01 = wmma_f32(a0, b1, c01);
    c10 = wmma_f32(a1, b0, c10);
    c11 = wmma_f32(a1, b1, c11);
  }

  // C/D layout: VGPR r -> M = r (lanes 0-15) or r+8 (lanes 16-31), N = idx
  const int mo = (lane >> 4) << 3;
  #pragma unroll
  for (int r = 0; r < 8; ++r) {
    const int m = mo + r;
    Pp[(size_t)(i0 + m) * MM + (j0 + idx)]           = c00[r];
    Pp[(size_t)(i0 + m) * MM + (j0 + 16 + idx)]      = c01[r];
    Pp[(size_t)(i0 + 16 + m) * MM + (j0 + idx)]      = c10[r];
    Pp[(size_t)(i0 + 16 + m) * MM + (j0 + 16 + idx)] = c11[r];
  }
}

// ---------------------------------------------------------------------------
// Deterministic fixed-order reduction of K-partials, fused with the
// Newton-Schulz epilogue:  W = 1.5*I - 0.5*G
// ---------------------------------------------------------------------------
__global__ __launch_bounds__(256) void w_reduce_kernel(
    const float* __restrict__ P, float* __restrict__ W, int nsplit) {
  const int e = blockIdx.x * 256 + threadIdx.x;   // 0 .. MM*MM-1
  float s = 0.0f;
  for (int p = 0; p < nsplit; ++p) s += P[(size_t)p * MM * MM + e];
  const int i = e >> 9;            // MM == 512
  const int j = e & (MM - 1);
  W[e] = ((i == j) ? 1.5f : 0.0f) - 0.5f * s;
}

// ---------------------------------------------------------------------------
// Y = X * W   ([DD,MM] x [MM,MM]).  Each wave owns a 32x32 output block
// (2x2 WMMA tiles): 4 independent accumulator chains for WMMA ILP and 2x
// operand reuse.  A-operands are contiguous K-pairs per lane -> b64 loads.
// ---------------------------------------------------------------------------
__global__ __launch_bounds__(128) void update_kernel(
    const float* __restrict__ X, const float* __restrict__ W,
    float* __restrict__ Y) {
  const int wave = threadIdx.x >> 5;
  const int lane = threadIdx.x & 31;
  const int tile = blockIdx.x * 4 + wave;       // 0..1023  (64x16 blocks of 32x32)
  const int i0 = (tile >> 4) * 32;
  const int j0 = (tile & 15) * 32;
  const int idx = lane & 15;
  const int kb  = (lane >> 4) << 1;             // 0 or 2
  const float* __restrict__ Xrow0 = X + (size_t)(i0 + idx) * MM;       // rows i0..+15
  const float* __restrict__ Xrow1 = X + (size_t)(i0 + 16 + idx) * MM;  // rows i0+16..+31

  v8f c00 = {}, c01 = {}, c10 = {}, c11 = {};
  #pragma unroll 4
  for (int k = 0; k < MM; k += 4) {
    v2f a0 = *reinterpret_cast<const v2f*>(Xrow0 + k + kb);  // K=kb,kb+1 contiguous
    v2f a1 = *reinterpret_cast<const v2f*>(Xrow1 + k + kb);
    const float* Wk0 = W + (size_t)(k + kb) * MM;
    const float* Wk1 = Wk0 + MM;
    v2f b0, b1;
    b0.x = Wk0[j0 + idx];       b0.y = Wk1[j0 + idx];        // cols j0..j0+15
    b1.x = Wk0[j0 + 16 + idx];  b1.y = Wk1[j0 + 16 + idx];   // cols j0+16..j0+31
    c00 = wmma_f32(a0, b0, c00);
    c01 = wmma_f32(a0, b1, c01);
    c10 = wmma_f32(a1, b0, c10);
    c11 = wmma_f32(a1, b1, c11);
  }

  const int mo = (lane >> 4) << 3;
  #pragma unroll
  for (int r = 0; r < 8; ++r) {
    const int m = mo + r;
    Y[(size_t)(i0 + m) * MM + (j0 + idx)]           = c00[r];
    Y[(size_t)(i0 + m) * MM + (j0 + 16 + idx)]      = c01[r];
    Y[(size_t)(i0 + 16 + m) * MM + (j0 + idx)]      = c10[r];
    Y[(size_t)(i0 + 16 + m) * MM + (j0 + 16 + idx)] = c11[r];
  }
}

// ---------------------------------------------------------------------------
// Host launcher: fixed 20 Newton-Schulz steps (once converged the iteration
// is a stable fixed point, so dropping the data-dependent exit is
// result-equivalent and graph-capture safe). Ping-pong between d_ws and
// d_out; step 19 (odd) lands in d_out.
// ---------------------------------------------------------------------------
extern "C" void kernel_launch(void* const* d_in, const int* in_sizes, int n_in,
                              void* d_out, int out_size, void* d_ws, size_t ws_size,
                              hipStream_t stream) {
  (void)in_sizes; (void)n_in; (void)out_size;
  const float* x   = (const float*)d_in[0];
  float* out       = (float*)d_out;
  float* ping      = (float*)d_ws;                 // DD*MM floats (4 MB)
  float* W         = ping + (size_t)DD * MM;       // MM*MM floats (1 MB)
  float* P         = W + (size_t)MM * MM;          // S * MM*MM floats

  // Pick the largest split-K factor that fits the workspace (deterministic).
  int S = 1;
  for (int c = 8; c >= 1; c >>= 1) {
    size_t need = ((size_t)DD * MM + (size_t)MM * MM + (size_t)c * MM * MM)
                  * sizeof(float);
    if (ws_size >= need) { S = c; break; }
  }
  const int chunk = DD / S;

  const float* cur = x;
  for (int s = 0; s < NSTEPS; ++s) {
    gram_partial_kernel<<<dim3(64, S), 128, 0, stream>>>(cur, P, chunk);
    w_reduce_kernel<<<(MM * MM) / 256, 256, 0, stream>>>(P, W, S);
    float* dst = (s & 1) ? out : ping;            // s=19 (odd) -> d_out
    update_kernel<<<1024, 128, 0, stream>>>(cur, W, dst);
    cur = dst;
  }
}